// PixelDSNTDistanceDoublePredict_old_20083267076473
// MI455X (gfx1250) — compile-verified
//
#include <hip/hip_runtime.h>
#include <math.h>

// Problem constants (B, K, H, W) = (64, 2, 512, 512), fp32.
#define B_    64
#define K_    2
#define H_    512
#define W_    512
#define HW    (H_ * W_)        // 262144 elements per heatmap
#define NMAP  (B_ * K_)        // 128 heatmaps
#define NBLK  64               // blocks per heatmap
#define CHUNK (HW / NBLK)      // 4096 elements per block
#define TPB   256
#define F4PT  (CHUNK / 4 / TPB) // 4 float4 per thread per tensor

// ---- CDNA5 async global->LDS path (builtin confirmed present) -------------
#if defined(__has_builtin)
#  if __has_builtin(__builtin_amdgcn_global_load_async_to_lds_b128)
#    define USE_ASYNC 1
#  endif
#endif

#ifdef USE_ASYNC
// Builtin signature (from clang diagnostic): param0 is
//   '__attribute__((__vector_size__(4*sizeof(int)))) int __device__ *'
// i.e. a GCC-vector int4 pointer in AS1; param1 analogously in AS3 (LDS).
typedef int v4i_t __attribute__((vector_size(16)));
typedef __attribute__((address_space(1))) v4i_t g_v4i;
typedef __attribute__((address_space(3))) v4i_t l_v4i;

// Generic global pointers share numeric representation with AS1; generic LDS
// pointers carry the LDS byte offset in the low 32 bits (ISA aperture rule),
// so integer round-trips give legal AS-qualified pointers without relying on
// addrspacecast legality in C-style casts.
__device__ __forceinline__ g_v4i* gptr(const void* p) {
    return (g_v4i*)(unsigned long long)(uintptr_t)p;
}
__device__ __forceinline__ l_v4i* lptr(const void* p) {
    return (l_v4i*)(unsigned int)(uintptr_t)p;
}
#endif

// Online-softmax pair combine: (M; s,sx,sy scaled by exp(m-M)).
__device__ __forceinline__ void combine(float& m1, float& s1, float& sx1, float& sy1,
                                        float m2, float s2, float sx2, float sy2) {
    float M  = fmaxf(m1, m2);
    float c1 = __expf(m1 - M);
    float c2 = __expf(m2 - M);
    s1  = s1  * c1 + s2  * c2;
    sx1 = sx1 * c1 + sx2 * c2;
    sy1 = sy1 * c1 + sy2 * c2;
    m1  = M;
}

// ---------------------------------------------------------------------------
// Kernel 1: per-(heatmap, chunk) online softmax moments + target argmax.
// Grid: (NBLK, NMAP), 256 threads. Single pass over all bytes.
// ---------------------------------------------------------------------------
__global__ __launch_bounds__(TPB) void dsnt_partial(const float* __restrict__ inp,
                                                    const float* __restrict__ tgt,
                                                    float* __restrict__ ws) {
    const int blk = blockIdx.x;   // chunk within heatmap
    const int mi  = blockIdx.y;   // heatmap index = b*K + k
    const int tid = threadIdx.x;

    const size_t base = (size_t)mi * HW + (size_t)blk * CHUNK;
    const float4* in4 = (const float4*)(inp + base);
    const float4* tg4 = (const float4*)(tgt + base);

#ifdef USE_ASYNC
    __shared__ float4 lin[CHUNK / 4];   // 16 KB
    __shared__ float4 ltg[CHUNK / 4];   // 16 KB
    #pragma unroll
    for (int i = 0; i < F4PT; ++i) {
        const int slot = i * TPB + tid;
        __builtin_amdgcn_global_load_async_to_lds_b128(gptr(in4 + slot), lptr(&lin[slot]), 0, 0);
        __builtin_amdgcn_global_load_async_to_lds_b128(gptr(tg4 + slot), lptr(&ltg[slot]), 0, 0);
    }
    asm volatile("s_wait_asynccnt 0" ::: "memory");
    // Each thread reads back exactly the LDS slots it issued, so the per-wave
    // asynccnt wait is sufficient (no cross-wave barrier needed).
#endif

    float m = -INFINITY, s = 0.f, sx = 0.f, sy = 0.f;   // online softmax state
    float tv = -INFINITY; int ti = 0x7fffffff;          // target argmax state

    #pragma unroll
    for (int i = 0; i < F4PT; ++i) {
        const int slot = i * TPB + tid;
#ifdef USE_ASYNC
        const float4 v = lin[slot];
        const float4 t = ltg[slot];
#else
        const float4 v = in4[slot];
        const float4 t = tg4[slot];
#endif
        const int   g0   = blk * CHUNK + slot * 4;        // element index in map
        const float rowf = (float)((g0 >> 9) + 1);        // W == 512 == 2^9
        const float colf = (float)((g0 & 511) + 1);       // float4 never crosses a row
        const float vv[4] = {v.x, v.y, v.z, v.w};
        const float tt[4] = {t.x, t.y, t.z, t.w};
        #pragma unroll
        for (int j = 0; j < 4; ++j) {
            // branchless online-softmax update (2x v_exp_f32 per element)
            const float val = vv[j];
            const float mn  = fmaxf(m, val);
            const float c   = __expf(m - mn);   // first iter: exp(-inf)=0 zeroes state
            const float e   = __expf(val - mn);
            s  = s  * c + e;
            sx = sx * c + e * (colf + (float)j);
            sy = sy * c + e * rowf;
            m  = mn;
            // first-occurrence argmax (larger value wins; tie -> smaller index)
            const float ta = tt[j];
            const int   gi = g0 + j;
            if (ta > tv || (ta == tv && gi < ti)) { tv = ta; ti = gi; }
        }
    }

    __shared__ float r_m[TPB], r_s[TPB], r_sx[TPB], r_sy[TPB], r_tv[TPB];
    __shared__ int   r_ti[TPB];
    r_m[tid] = m; r_s[tid] = s; r_sx[tid] = sx; r_sy[tid] = sy;
    r_tv[tid] = tv; r_ti[tid] = ti;
    __syncthreads();
    for (int off = TPB / 2; off > 0; off >>= 1) {
        if (tid < off) {
            float cm = r_m[tid], cs = r_s[tid], cx = r_sx[tid], cy = r_sy[tid];
            combine(cm, cs, cx, cy, r_m[tid + off], r_s[tid + off], r_sx[tid + off], r_sy[tid + off]);
            r_m[tid] = cm; r_s[tid] = cs; r_sx[tid] = cx; r_sy[tid] = cy;
            const float tv2 = r_tv[tid + off]; const int ti2 = r_ti[tid + off];
            if (tv2 > r_tv[tid] || (tv2 == r_tv[tid] && ti2 < r_ti[tid])) {
                r_tv[tid] = tv2; r_ti[tid] = ti2;
            }
        }
        __syncthreads();
    }
    if (tid == 0) {
        float* w = ws + (size_t)(mi * NBLK + blk) * 8;   // 32B-aligned record
        w[0] = r_m[0]; w[1] = r_s[0]; w[2] = r_sx[0]; w[3] = r_sy[0];
        w[4] = r_tv[0]; ((int*)w)[5] = r_ti[0];
    }
}

// ---------------------------------------------------------------------------
// Kernel 2: merge NBLK partials per heatmap -> pred/true pixel coords.
// Output layout: out[0..3] scalars, out[4..259] pred_pix-1, out[260..515] true_pix-1.
// ---------------------------------------------------------------------------
__global__ __launch_bounds__(NBLK) void dsnt_combine_k(const float* __restrict__ ws,
                                                       float* __restrict__ out) {
    const int mi  = blockIdx.x;
    const int tid = threadIdx.x;   // 0..63
    const float* w = ws + (size_t)(mi * NBLK + tid) * 8;

    __shared__ float r_m[NBLK], r_s[NBLK], r_sx[NBLK], r_sy[NBLK], r_tv[NBLK];
    __shared__ int   r_ti[NBLK];
    r_m[tid] = w[0]; r_s[tid] = w[1]; r_sx[tid] = w[2]; r_sy[tid] = w[3];
    r_tv[tid] = w[4]; r_ti[tid] = ((const int*)w)[5];
    __syncthreads();
    for (int off = NBLK / 2; off > 0; off >>= 1) {
        if (tid < off) {
            float cm = r_m[tid], cs = r_s[tid], cx = r_sx[tid], cy = r_sy[tid];
            combine(cm, cs, cx, cy, r_m[tid + off], r_s[tid + off], r_sx[tid + off], r_sy[tid + off]);
            r_m[tid] = cm; r_s[tid] = cs; r_sx[tid] = cx; r_sy[tid] = cy;
            const float tv2 = r_tv[tid + off]; const int ti2 = r_ti[tid + off];
            if (tv2 > r_tv[tid] || (tv2 == r_tv[tid] && ti2 < r_ti[tid])) {
                r_tv[tid] = tv2; r_ti[tid] = ti2;
            }
        }
        __syncthreads();
    }
    if (tid == 0) {
        // pred_x = Σ p·(col+1)  (already pixel-scaled: x_grid*W == col+1)
        const float predx = r_sx[0] / r_s[0];
        const float predy = r_sy[0] / r_s[0];
        const int   idx   = r_ti[0];
        const float truex = (float)((idx & 511) + 1);
        const float truey = (float)((idx >> 9) + 1);
        out[4 + mi * 2 + 0]       = predx - 1.0f;
        out[4 + mi * 2 + 1]       = predy - 1.0f;
        out[4 + 256 + mi * 2 + 0] = truex - 1.0f;
        out[4 + 256 + mi * 2 + 1] = truey - 1.0f;
    }
}

// ---------------------------------------------------------------------------
// Kernel 3: scalar losses. Stored coords are (pix-1); all terms are
// differences, so the -1 offsets cancel exactly.
// ---------------------------------------------------------------------------
__global__ __launch_bounds__(B_) void dsnt_final(float* __restrict__ out) {
    const int b = threadIdx.x;   // 0..63
    const float* pp = out + 4;
    const float* tp = out + 4 + 256;
    const int m0 = (b * 2 + 0) * 2;   // keypoint 0 (inferior)
    const int m1 = (b * 2 + 1) * 2;   // keypoint 1 (superior)
    const float px0 = pp[m0], py0 = pp[m0 + 1], px1 = pp[m1], py1 = pp[m1 + 1];
    const float tx0 = tp[m0], ty0 = tp[m0 + 1], tx1 = tp[m1], ty1 = tp[m1 + 1];

    const float ed0 = sqrtf((tx0 - px0) * (tx0 - px0) + (ty0 - py0) * (ty0 - py0));
    const float ed1 = sqrtf((tx1 - px1) * (tx1 - px1) + (ty1 - py1) * (ty1 - py1));
    const float pd  = sqrtf((px0 - px1) * (px0 - px1) + (py0 - py1) * (py0 - py1));
    const float td  = sqrtf((tx0 - tx1) * (tx0 - tx1) + (ty0 - ty1) * (ty0 - ty1));

    __shared__ float a0[B_], a1[B_], a3[B_];
    a0[b] = ed0; a1[b] = ed1; a3[b] = pd - td;
    __syncthreads();
    for (int off = B_ / 2; off > 0; off >>= 1) {
        if (b < off) { a0[b] += a0[b + off]; a1[b] += a1[b + off]; a3[b] += a3[b + off]; }
        __syncthreads();
    }
    if (b == 0) {
        const float si = a0[0] / (float)B_;
        const float ss = a1[0] / (float)B_;
        out[0] = si;
        out[1] = ss;
        out[2] = si + ss;
        out[3] = a3[0] / (float)B_;
    }
}

extern "C" void kernel_launch(void* const* d_in, const int* in_sizes, int n_in,
                              void* d_out, int out_size, void* d_ws, size_t ws_size,
                              hipStream_t stream) {
    const float* inp = (const float*)d_in[0];
    const float* tgt = (const float*)d_in[1];
    float* out = (float*)d_out;
    float* ws  = (float*)d_ws;   // needs NMAP*NBLK*8*4 = 256 KB

    dim3 gridA(NBLK, NMAP);
    dsnt_partial<<<gridA, TPB, 0, stream>>>(inp, tgt, ws);
    dsnt_combine_k<<<NMAP, NBLK, 0, stream>>>(ws, out);
    dsnt_final<<<1, B_, 0, stream>>>(out);
}